// Sgcn_32418413150849
// MI455X (gfx1250) — compile-verified
//
#include <hip/hip_runtime.h>
#include <hip/hip_bf16.h>

typedef __attribute__((ext_vector_type(2))) float v2f;
typedef __attribute__((ext_vector_type(8))) float v8f;

#define V_NODES 500
#define V_PAD 512                              // padded M/K for the hop GEMMs
#define L_LEN 64
#define C_CH 64
#define N_BATCH 32
#define E_EMB 10
#define VL ((size_t)V_NODES * L_LEN)           // 32000
#define NCVL ((size_t)N_BATCH * C_CH * VL)     // 65,536,000 floats

// ---------------------------------------------------------------------------
// zero-fill (for padded adjacency buffers)
// ---------------------------------------------------------------------------
__global__ __launch_bounds__(256) void zero_kernel(float* __restrict__ p, int n) {
  int i = blockIdx.x * blockDim.x + threadIdx.x;
  if (i < n) p[i] = 0.f;
}

// ---------------------------------------------------------------------------
// Embedding affine update: nv1' = nv1 @ et_w + et_b ; nv2' = (nv2^T @ et_w + et_b)^T
// ---------------------------------------------------------------------------
__global__ __launch_bounds__(256) void emb_kernel(
    const float* __restrict__ nv1, const float* __restrict__ nv2,
    const float* __restrict__ etw, const float* __restrict__ etb,
    float* __restrict__ nv1b, float* __restrict__ nv2b) {
  int i = blockIdx.x * blockDim.x + threadIdx.x;
  if (i < V_NODES * E_EMB) {                       // nv1b[v][e]
    int v = i / E_EMB, e = i % E_EMB;
    float acc = etb[e];
#pragma unroll
    for (int f = 0; f < E_EMB; ++f) acc += nv1[v * E_EMB + f] * etw[f * E_EMB + e];
    nv1b[i] = acc;
  } else if (i < 2 * V_NODES * E_EMB) {            // nv2b[e][w]
    int j = i - V_NODES * E_EMB;
    int e = j / V_NODES, w = j % V_NODES;
    float acc = etb[e];
#pragma unroll
    for (int f = 0; f < E_EMB; ++f) acc += nv2[f * V_NODES + w] * etw[f * E_EMB + e];
    nv2b[e * V_NODES + w] = acc;
  }
}

// ---------------------------------------------------------------------------
// adpT[w][v] = softmax_w(relu(nv1[v,:] @ nv2[:,w])), stored transposed with
// row stride V_PAD (pad region pre-zeroed). One block per row v; 512 threads.
// ---------------------------------------------------------------------------
__global__ __launch_bounds__(512) void adj_kernel(
    const float* __restrict__ nv1, const float* __restrict__ nv2,
    float* __restrict__ adpT) {
  const int v = blockIdx.x;
  const int w = threadIdx.x;
  const int lane = threadIdx.x & 31;
  const int waveId = threadIdx.x >> 5;
  __shared__ float e1[E_EMB];
  __shared__ float red[16];
  if (threadIdx.x < E_EMB) e1[threadIdx.x] = nv1[v * E_EMB + threadIdx.x];
  __syncthreads();

  const bool ok = (w < V_NODES);
  float s = 0.f;
  if (ok) {
    float acc = 0.f;
#pragma unroll
    for (int e = 0; e < E_EMB; ++e) acc += e1[e] * nv2[e * V_NODES + w];
    s = fmaxf(acc, 0.f);
  }
  // block max
  float m = ok ? s : -3.4e38f;
#pragma unroll
  for (int off = 16; off > 0; off >>= 1) m = fmaxf(m, __shfl_xor(m, off, 32));
  if (lane == 0) red[waveId] = m;
  __syncthreads();
  if (threadIdx.x < 16) {
    float t = red[threadIdx.x];
#pragma unroll
    for (int off = 8; off > 0; off >>= 1) t = fmaxf(t, __shfl_xor(t, off, 32));
    red[threadIdx.x] = t;
  }
  __syncthreads();
  m = red[0];
  float ex = ok ? __expf(s - m) : 0.f;
  __syncthreads();                      // done reading red before reuse
  // block sum
  float sum = ex;
#pragma unroll
  for (int off = 16; off > 0; off >>= 1) sum += __shfl_xor(sum, off, 32);
  if (lane == 0) red[waveId] = sum;
  __syncthreads();
  if (threadIdx.x < 16) {
    float t = red[threadIdx.x];
#pragma unroll
    for (int off = 8; off > 0; off >>= 1) t += __shfl_xor(t, off, 32);
    red[threadIdx.x] = t;
  }
  __syncthreads();
  sum = red[0];
  if (ok) adpT[(size_t)w * V_PAD + v] = ex / sum;
}

// ---------------------------------------------------------------------------
// Hop GEMM: per (n,c): Y[w,l] = sum_v adpT[w,v] * Xsh[v,l],
//   Xsh[v,l] = (l >= SHIFT) ? X[v, l-SHIFT] : 0
// grid = (8, N*C); block = 128 (4 waves, one 16-row M tile each, all four
// 16-col N tiles). X chunks (32x64) staged transposed in LDS so B fragments
// are single ds_load_b64; A fragments are global_load_b64 from padded adpT.
// Branch-free inner loop: 8x(1 b64 A + 4 b64 B-ds + 4 wmma) per chunk.
// ---------------------------------------------------------------------------
template <int SHIFT>
__global__ __launch_bounds__(128) void hop_kernel(
    const float* __restrict__ adpT,    // [V_PAD][V_PAD], zero-padded
    const float* __restrict__ Xin,     // [NC][500][64]
    float* __restrict__ Yout) {        // [NC][500][64]
  __shared__ float xt[64][34];         // transposed chunk, stride 34 (8B-align + bank spread)

  const int tid = threadIdx.x;
  const int lane = tid & 31;
  const int wave = tid >> 5;
  const int mBase = (blockIdx.x * 4 + wave) * 16;   // 0..496
  const size_t nc = blockIdx.y;
  const float* X = Xin + nc * VL;
  float* Y = Yout + nc * VL;

  const int half = lane >> 4;          // 0 | 1
  const int lid = lane & 15;
  const int kOff = half * 2;           // K sub-offset of this lane's A/B pair
  const float* arow = adpT + (size_t)(mBase + lid) * V_PAD;

  // staging role: thread -> (row tid>>2 of chunk, 16 cols starting (tid&3)*16)
  const int sRow = tid >> 2;           // 0..31
  const int sCol = (tid & 3) * 16;     // 0,16,32,48

  v8f acc[4] = {v8f{}, v8f{}, v8f{}, v8f{}};

  for (int kc = 0; kc < V_PAD; kc += 32) {
    __syncthreads();                   // previous chunk fully consumed
    {
      const int krow = kc + sRow;
      float vals[16];
      if (krow < V_NODES) {
        const float4* src = (const float4*)(X + (size_t)krow * L_LEN + sCol);
#pragma unroll
        for (int q = 0; q < 4; ++q) {
          float4 f = src[q];
          vals[q * 4 + 0] = f.x; vals[q * 4 + 1] = f.y;
          vals[q * 4 + 2] = f.z; vals[q * 4 + 3] = f.w;
        }
      } else {
#pragma unroll
        for (int q = 0; q < 16; ++q) vals[q] = 0.f;
      }
#pragma unroll
      for (int q = 0; q < 16; ++q) {
        const int dcol = sCol + q + SHIFT;      // roll along l
        if (dcol < L_LEN) xt[dcol][sRow] = vals[q];
      }
      if (SHIFT > 0 && sCol == 0) {
#pragma unroll
        for (int q = 0; q < SHIFT; ++q) xt[q][sRow] = 0.f;  // zero-masked head
      }
    }
    __syncthreads();

#pragma unroll
    for (int ks = 0; ks < 8; ++ks) {
      const v2f a = *(const v2f*)(arow + kc + ks * 4 + kOff);
#pragma unroll
      for (int t = 0; t < 4; ++t) {
        const v2f b = *(const v2f*)(&xt[t * 16 + lid][ks * 4 + kOff]);
        acc[t] = __builtin_amdgcn_wmma_f32_16x16x4_f32(
            false, a, false, b, (short)0, acc[t], false, false);
      }
    }
  }

  // C/D layout: VGPR r -> M = r (lanes 0-15) / 8+r (lanes 16-31); N = lid
#pragma unroll
  for (int t = 0; t < 4; ++t) {
    const int l = t * 16 + lid;
#pragma unroll
    for (int r = 0; r < 8; ++r) {
      const int mm = mBase + half * 8 + r;
      if (mm < V_NODES) Y[(size_t)mm * L_LEN + l] = acc[t][r];
    }
  }
}

// ---------------------------------------------------------------------------
// MLP (1x1 conv): per n: out[o, col] = sum_{k<192} W[o,k] * H[k, col] + b[o]
// H rows 0..63 = x, 64..127 = x1, 128..191 = x2. Three compile-time sub-loops
// avoid dynamic buffer selection; W fragments are b64.
// ---------------------------------------------------------------------------
__device__ __forceinline__ v8f mlp_part(v8f acc, const float* __restrict__ wseg,
                                        const float* __restrict__ h, int col,
                                        int kOff) {
#pragma unroll
  for (int kb = 0; kb < C_CH; kb += 4) {
    const v2f a = *(const v2f*)(wseg + kb + kOff);
    v2f b;
    b.x = h[(size_t)(kb + kOff) * VL + col];
    b.y = h[(size_t)(kb + kOff + 1) * VL + col];
    acc = __builtin_amdgcn_wmma_f32_16x16x4_f32(
        false, a, false, b, (short)0, acc, false, false);
  }
  return acc;
}

__global__ __launch_bounds__(128) void mlp_kernel(
    const float* __restrict__ x, const float* __restrict__ x1,
    const float* __restrict__ x2, const float* __restrict__ W,
    const float* __restrict__ bias, float* __restrict__ out) {
  const int lane = threadIdx.x & 31;
  const int wave = threadIdx.x >> 5;           // o-tile
  const int oBase = wave * 16;
  const int n = blockIdx.y;
  const int half = lane >> 4;
  const int lid = lane & 15;
  const int kOff = half * 2;
  const int col = blockIdx.x * 16 + lid;

  const size_t nOff = (size_t)n * C_CH * VL;
  const float* wrow = W + (size_t)(oBase + lid) * (3 * C_CH);

  v8f acc = {};
  acc = mlp_part(acc, wrow, x + nOff, col, kOff);
  acc = mlp_part(acc, wrow + C_CH, x1 + nOff, col, kOff);
  acc = mlp_part(acc, wrow + 2 * C_CH, x2 + nOff, col, kOff);

  float* O = out + nOff;
#pragma unroll
  for (int r = 0; r < 8; ++r) {
    const int o = oBase + half * 8 + r;
    O[(size_t)o * VL + col] = acc[r] + bias[o];
  }
}

// ---------------------------------------------------------------------------
extern "C" void kernel_launch(void* const* d_in, const int* in_sizes, int n_in,
                              void* d_out, int out_size, void* d_ws, size_t ws_size,
                              hipStream_t stream) {
  const float* x   = (const float*)d_in[0];
  const float* nv1 = (const float*)d_in[1];
  const float* nv2 = (const float*)d_in[2];
  const float* etw = (const float*)d_in[3];
  const float* etb = (const float*)d_in[4];
  const float* mw  = (const float*)d_in[5];
  const float* mb  = (const float*)d_in[6];
  float* out = (float*)d_out;

  // workspace carve-up
  char* ws = (char*)d_ws;
  size_t off = 0;
  float* adpT0 = (float*)(ws + off); off += (size_t)V_PAD * V_PAD * 4;
  float* adpT1 = (float*)(ws + off); off += (size_t)V_PAD * V_PAD * 4;
  float* nv1b  = (float*)(ws + off); off += (size_t)V_NODES * E_EMB * 4;
  float* nv2b  = (float*)(ws + off); off += (size_t)V_NODES * E_EMB * 4;
  off = (off + 255) & ~(size_t)255;
  float* x1 = (float*)(ws + off); off += NCVL * 4;
  float* x2 = (float*)(ws + off); off += NCVL * 4;

  // 0) zero the padded adjacency buffers (adjacent -> single launch)
  const int adjElems = 2 * V_PAD * V_PAD;
  zero_kernel<<<(adjElems + 255) / 256, 256, 0, stream>>>(adpT0, adjElems);

  // 1) adjacency #0; 2) embedding update -> adjacency #1
  adj_kernel<<<V_NODES, 512, 0, stream>>>(nv1, nv2, adpT0);
  emb_kernel<<<(2 * V_NODES * E_EMB + 255) / 256, 256, 0, stream>>>(
      nv1, nv2, etw, etb, nv1b, nv2b);
  adj_kernel<<<V_NODES, 512, 0, stream>>>(nv1b, nv2b, adpT1);

  // 3) hop GEMMs (WMMA f32 16x16x4, LDS-staged B)
  dim3 gh(8, N_BATCH * C_CH);
  hop_kernel<0><<<gh, 128, 0, stream>>>(adpT0, x, x1);
  hop_kernel<2><<<gh, 128, 0, stream>>>(adpT1, x1, x2);

  // 4) fused channel-MLP + bias
  dim3 gm((int)(VL / 16), N_BATCH);
  mlp_kernel<<<gm, 128, 0, stream>>>(x, x1, x2, mw, mb, out);
}